// ReLUTransformer_7138235646107
// MI455X (gfx1250) — compile-verified
//
#include <hip/hip_runtime.h>

typedef __attribute__((ext_vector_type(2))) float v2f;
typedef __attribute__((ext_vector_type(4))) float v4f;
typedef __attribute__((ext_vector_type(8))) float v8f;

#define ZD     4096            // feature dimension d
#define ZNERR  8192            // number of error rows
#define ZCHUNK 32              // row chunks for partial reduction
#define ZROWS  (ZNERR / ZCHUNK) // 256 rows per chunk

// ---------------------------------------------------------------------------
// Kernel 1: column-wise abs-sum reduction via V_WMMA_F32_16X16X4_F32.
// A = all-ones (16x4), B = |e| tile (4x16)  =>  D[m][n] = sum_k |e| (all m equal).
// Each wave owns 64 columns and 256 rows; 4 independent v8f accumulators,
// one per column sub-slot j in {0..3}; lane n (n = lane&15) streams float4s
// covering columns c0+4n..c0+4n+3, rows r+half and r+2+half (half = lane>>4).
// Because A is all ones, the sum is invariant to the exact K-slot layout of
// the B operand; only the N<->lane mapping matters, and we store results with
// the same mapping. Loads use default RT policy so `e` (128 MB) is resident
// in the 192 MB L2 for the second pass (kernel 3).
// ---------------------------------------------------------------------------
__global__ __launch_bounds__(256) void zono_colabs_wmma(
    const float* __restrict__ e, float* __restrict__ partials) {
  const int lane = threadIdx.x & 31;
  const int wave = threadIdx.x >> 5;
  const int n    = lane & 15;
  const int half = lane >> 4;
  const int c0   = (blockIdx.x * 8 + wave) * 64;   // 64 columns per wave
  const int row0 = blockIdx.y * ZROWS;             // 256 rows per chunk

  v2f ones; ones.x = 1.0f; ones.y = 1.0f;
  v8f acc0 = {}; v8f acc1 = {}; v8f acc2 = {}; v8f acc3 = {};

  const float* base = e + (size_t)row0 * ZD + c0 + 4 * n;
  #pragma unroll 4
  for (int it = 0; it < ZROWS / 4; ++it) {
    const float* p = base + (size_t)(it * 4 + half) * ZD;
    v4f q1 = *(const v4f*)(p);
    v4f q2 = *(const v4f*)(p + 2 * (size_t)ZD);
    q1.x = __builtin_fabsf(q1.x); q1.y = __builtin_fabsf(q1.y);
    q1.z = __builtin_fabsf(q1.z); q1.w = __builtin_fabsf(q1.w);
    q2.x = __builtin_fabsf(q2.x); q2.y = __builtin_fabsf(q2.y);
    q2.z = __builtin_fabsf(q2.z); q2.w = __builtin_fabsf(q2.w);

    v2f b0; b0.x = q1.x; b0.y = q2.x;
    v2f b1; b1.x = q1.y; b1.y = q2.y;
    v2f b2; b2.x = q1.z; b2.y = q2.z;
    v2f b3; b3.x = q1.w; b3.y = q2.w;

    acc0 = __builtin_amdgcn_wmma_f32_16x16x4_f32(false, ones, false, b0,
                                                 (short)0, acc0, false, false);
    acc1 = __builtin_amdgcn_wmma_f32_16x16x4_f32(false, ones, false, b1,
                                                 (short)0, acc1, false, false);
    acc2 = __builtin_amdgcn_wmma_f32_16x16x4_f32(false, ones, false, b2,
                                                 (short)0, acc2, false, false);
    acc3 = __builtin_amdgcn_wmma_f32_16x16x4_f32(false, ones, false, b3,
                                                 (short)0, acc3, false, false);
  }

  // C/D layout: VGPR0 @ lanes 0-15 = row M=0, N=lane. All M rows identical.
  if (lane < 16) {
    v4f out; out.x = acc0[0]; out.y = acc1[0]; out.z = acc2[0]; out.w = acc3[0];
    *(v4f*)(partials + (size_t)blockIdx.y * ZD + c0 + 4 * n) = out;
  }
}

// ---------------------------------------------------------------------------
// Kernel 2: fold 32 partials -> apt, then the per-column zonotope ReLU math.
// ---------------------------------------------------------------------------
__global__ __launch_bounds__(256) void zono_finalize(
    const float* __restrict__ partials, const float* __restrict__ center,
    float* __restrict__ out_center, float* __restrict__ scale,
    float* __restrict__ mu) {
  const int c = blockIdx.x * blockDim.x + threadIdx.x;
  if (c >= ZD) return;
  float apt = 0.0f;
  #pragma unroll
  for (int k = 0; k < ZCHUNK; ++k) apt += partials[k * ZD + c];
  const float cc = center[c];
  const float ub = cc + apt;
  const float lb = cc - apt;
  const bool cross = (ub > 0.0f) && (lb < 0.0f);
  const bool act   = (lb >= 0.0f);
  const float denom = cross ? (ub - lb) : 1.0f;
  const float slope = cross ? (ub / denom) : 0.0f;
  const float m     = cross ? (-slope * lb * 0.5f) : 0.0f;
  out_center[c] = act ? cc : (slope * cc + m);
  scale[c]      = act ? 1.0f : slope;
  mu[c]         = m;
}

// ---------------------------------------------------------------------------
// Kernel 3: top = e * scale[col], streamed as float4.
//  - reads of `e` use default RT policy -> hit L2 (populated by kernel 1)
//  - output stores are NON-TEMPORAL so the 128 MB of writes do not evict `e`
//    (or anything else) from L2; output is never re-read.
//  - gfx1250 global_prefetch_b8 one stride ahead.
// ---------------------------------------------------------------------------
__global__ __launch_bounds__(256) void zono_scale_top(
    const float* __restrict__ e, const float* __restrict__ scale,
    float* __restrict__ out_top) {
  const int C4 = ZD / 4;                       // 1024 float4 per row
  const long total = (long)ZNERR * C4;         // 8,388,608 float4
  const long stride = (long)gridDim.x * blockDim.x;
  for (long idx = (long)blockIdx.x * blockDim.x + threadIdx.x; idx < total;
       idx += stride) {
    const long c4 = idx & (C4 - 1);
    v4f v = ((const v4f*)e)[idx];
    v4f s = ((const v4f*)scale)[c4];
    __builtin_prefetch(e + 4 * (idx + stride), 0, 0);  // global_prefetch_b8
    v.x *= s.x; v.y *= s.y; v.z *= s.z; v.w *= s.w;
    __builtin_nontemporal_store(v, ((v4f*)out_top) + idx);  // TH=NT store
  }
}

// ---------------------------------------------------------------------------
// Kernel 4: bottom = diag(mu); float4 zero stream + diagonal insert.
// Stores are non-temporal (64 MB of zeros must not pollute L2).
// ---------------------------------------------------------------------------
__global__ __launch_bounds__(256) void zono_diag_bottom(
    const float* __restrict__ mu, float* __restrict__ out_bottom) {
  const int C4 = ZD / 4;
  const long total = (long)ZD * C4;            // 4,194,304 float4
  const long stride = (long)gridDim.x * blockDim.x;
  for (long idx = (long)blockIdx.x * blockDim.x + threadIdx.x; idx < total;
       idx += stride) {
    const int row = (int)(idx >> 10);          // idx / C4
    const int c4  = (int)(idx & (C4 - 1));
    v4f v = {0.0f, 0.0f, 0.0f, 0.0f};
    if (c4 == (row >> 2)) v[row & 3] = mu[row];
    __builtin_nontemporal_store(v, ((v4f*)out_bottom) + idx);  // TH=NT store
  }
}

extern "C" void kernel_launch(void* const* d_in, const int* in_sizes, int n_in,
                              void* d_out, int out_size, void* d_ws, size_t ws_size,
                              hipStream_t stream) {
  const float* center = (const float*)d_in[0];   // (4096,)
  const float* error  = (const float*)d_in[1];   // (8192, 4096)

  // Workspace layout: partials[32][4096] | scale[4096] | mu[4096]
  float* partials = (float*)d_ws;
  float* scale    = partials + (size_t)ZCHUNK * ZD;
  float* mu       = scale + ZD;

  // Output layout: new_center (4096) | top (8192x4096) | bottom (4096x4096)
  float* out_center = (float*)d_out;
  float* out_top    = out_center + ZD;
  float* out_bottom = out_top + (size_t)ZNERR * ZD;

  // K1: 8 column-blocks x 32 row-chunks, 8 waves/block (one 64-col tile each)
  zono_colabs_wmma<<<dim3(ZD / 512, ZCHUNK), 256, 0, stream>>>(error, partials);
  // K2: per-column math
  zono_finalize<<<ZD / 256, 256, 0, stream>>>(partials, center, out_center,
                                              scale, mu);
  // K3: stream-scale top rows (grid-stride, ~8 float4 per thread, NT stores)
  zono_scale_top<<<4096, 256, 0, stream>>>(error, scale, out_top);
  // K4: diagonal bottom block (grid-stride, ~8 float4 per thread, NT stores)
  zono_diag_bottom<<<2048, 256, 0, stream>>>(mu, out_bottom);
}